// SSMLoRALayer_66992899883693
// MI455X (gfx1250) — compile-verified
//
#include <hip/hip_runtime.h>
#include <stdint.h>

// fp32 WMMA fragment types for V_WMMA_F32_16X16X4_F32 (wave32):
//   A 16x4  f32 -> 2 VGPRs,  B 4x16 f32 -> 2 VGPRs,  C/D 16x16 f32 -> 8 VGPRs
typedef float v2f __attribute__((ext_vector_type(2)));
typedef float v8f __attribute__((ext_vector_type(8)));

#define D_IN   4096
#define D_OUT  4096
#define RNK    16
#define NROWS  (4 * 4096)      // B * S
#define ALPHA  32.0f
#define EPS    1e-8f
#define SLOPE  0.01f

#define KC     64                    // K floats staged per async chunk (256B/row)
#define NCHUNK ((D_IN / 8) / KC)     // 8 chunks per wave (wave K-slice = 512)
#define XPAD   68                    // padded LDS row stride in floats: 272B, 16B-aligned
#define BUFB   (16 * XPAD * 4)       // bytes per x staging buffer (one wave, one phase)

__global__ __launch_bounds__(256)
void ssm_lora_fused(const float* __restrict__ x,
                    const float* __restrict__ h_t,
                    const float* __restrict__ W_a,
                    const float* __restrict__ W_b,
                    const float* __restrict__ W_c,
                    const float* __restrict__ W_d,
                    float* __restrict__ y)
{
    __shared__ float sX[8][2][16][XPAD];   // per-wave double-buffered x tile (async dest)
    __shared__ float sWc[RNK][RNK];
    __shared__ float sWd[RNK][RNK];
    __shared__ float sHt[RNK][RNK + 1];
    __shared__ float sNx[RNK][RNK + 1];
    __shared__ float sH1[RNK][RNK + 1];
    __shared__ float sS [RNK][RNK + 1];
    __shared__ float sPart[8][8][32];      // [wave][acc vgpr][lane]

    const int t  = threadIdx.x;
    const int wv = t >> 5;               // wave id 0..7
    const int l  = t & 31;               // lane id
    const int m  = l & 15;               // M (GEMM1 A) / N (GEMM B) index
    const int c0 = (l < 16) ? 0 : 2;     // K sub-offset per half-wave (ISA fp32 frag layout)
    const int rowbase = blockIdx.x * 16;

    // ---- phase 0: stage tiny operands ----
    {
        const int r_ = t >> 4, j_ = t & 15;
        sWc[r_][j_] = W_c[t];
        sWd[r_][j_] = W_d[t];
        sHt[r_][j_] = h_t[(size_t)(rowbase + r_) * RNK + j_];
    }

    // ---- phase 1: new_x = x @ W_a ----
    // x tiles staged LDS-side via async global->LDS (coalesced b128, double-buffered on
    // ASYNCcnt); A fragments then read with ds_load_b64; W_a B-fragments from L2.
    v8f acc = {};
    {
        const int kbeg = wv * (D_IN / 8);

        // per-lane async-copy geometry: half-wave = one 256B row segment
        const int      r0 = l >> 4;           // row parity within a 2-row group
        const unsigned cl = (unsigned)(m * 4);// 16B column group (floats)
        unsigned gbase = ((unsigned)(rowbase + r0) * (unsigned)D_IN + (unsigned)kbeg + cl) * 4u;
        unsigned lbase = (unsigned)(uintptr_t)(&sX[wv][0][0][0]) + (r0 * XPAD + (int)cl) * 4u;

        // issue one 16x64 chunk (8 x b128 per lane-group) into buffer `buf`
        #define ASYNC_CHUNK(CIDX, BUF)                                              \
        {                                                                           \
            unsigned g_   = gbase + (unsigned)(CIDX) * (KC * 4u);                   \
            unsigned lds_ = lbase + (unsigned)(BUF) * (unsigned)BUFB;               \
            _Pragma("unroll")                                                       \
            for (int j_ = 0; j_ < 8; ++j_) {                                        \
                asm volatile("global_load_async_to_lds_b128 %0, %1, %2"             \
                             :: "v"(lds_), "v"(g_), "s"(x) : "memory");             \
                g_   += 2u * D_IN * 4u;                                             \
                lds_ += 2u * XPAD * 4u;                                             \
            }                                                                       \
        }

        ASYNC_CHUNK(0, 0);
        for (int c = 0; c < NCHUNK; ++c) {
            const int buf = c & 1;
            if (c + 1 < NCHUNK) {
                ASYNC_CHUNK(c + 1, buf ^ 1);
                asm volatile("s_wait_asynccnt 0x8" ::: "memory");  // chunk c complete
            } else {
                asm volatile("s_wait_asynccnt 0x0" ::: "memory");
            }
            const float* __restrict__ xb = &sX[wv][buf][m][0];
            const int kc = kbeg + c * KC;
            #pragma unroll
            for (int q = 0; q < 16; ++q) {
                // A fragment from LDS: x[row=m][kc+4q+c0 .. +1]  (ds_load_b64)
                v2f a = *(const v2f*)(xb + 4 * q + c0);
                // B fragment: W_a[k+c0][N=m], W_a[k+c0+1][N=m]  (L2-resident)
                const int k = kc + 4 * q;
                v2f b;
                b.x = W_a[(size_t)(k + c0)     * RNK + m];
                b.y = W_a[(size_t)(k + c0 + 1) * RNK + m];
                acc = __builtin_amdgcn_wmma_f32_16x16x4_f32(
                    false, a, false, b, (short)0, acc, false, false);
            }
        }
        #undef ASYNC_CHUNK
    }
    #pragma unroll
    for (int v = 0; v < 8; ++v) sPart[wv][v][l] = acc[v];
    __syncthreads();

    // cross-wave K reduction -> plain-layout new_x tile in LDS
    {
        const int v = t >> 5, lane = t & 31;
        float s = 0.f;
        #pragma unroll
        for (int w = 0; w < 8; ++w) s += sPart[w][v][lane];
        const int row = v + ((lane < 16) ? 0 : 8);   // C/D layout: vgpr v holds M=v / M=v+8
        sNx[row][lane & 15] = s;
    }
    __syncthreads();

    // ---- phase 2: recurrence + min-max norm + leaky relu (tiny, VALU) ----
    {
        const int row = t >> 4, j = t & 15;
        float hd = 0.f, nd = 0.f;
        #pragma unroll
        for (int k = 0; k < RNK; ++k) {
            hd = __builtin_fmaf(sHt[row][k], sWc[k][j], hd);
            nd = __builtin_fmaf(sNx[row][k], sWd[k][j], nd);
        }
        const float h1 = hd + nd + sHt[row][j];
        sH1[row][j] = h1;
        __syncthreads();
        float mn = sH1[row][0], mx = mn;
        #pragma unroll
        for (int k = 1; k < RNK; ++k) {
            const float vv = sH1[row][k];
            mn = fminf(mn, vv);
            mx = fmaxf(mx, vv);
        }
        const float hn  = (h1 - mn) / (mx - mn + EPS);
        const float act = (hn >= 0.f) ? hn : SLOPE * hn;
        sS[row][j] = sNx[row][j] + act;
    }
    __syncthreads();

    // ---- phase 3: y = ALPHA * (new_x + act) @ W_b  (N split over waves) ----
    v2f afrag[4];
    #pragma unroll
    for (int q = 0; q < 4; ++q) {       // K blocks 0,4,8,12
        afrag[q].x = sS[m][4 * q + c0];
        afrag[q].y = sS[m][4 * q + c0 + 1];
    }
    float* __restrict__ yrow0 = y + (size_t)rowbase * D_OUT;
    const int roff = (l < 16) ? 0 : 8;
    for (int i = 0; i < 32; ++i) {
        const int n0 = (i * 8 + wv) * 16;   // 8 waves cover 128 adjacent columns per i
        v8f o = {};
        #pragma unroll
        for (int q = 0; q < 4; ++q) {
            v2f b;
            b.x = W_b[(size_t)(4 * q + c0)     * D_OUT + n0 + m];
            b.y = W_b[(size_t)(4 * q + c0 + 1) * D_OUT + n0 + m];
            o = __builtin_amdgcn_wmma_f32_16x16x4_f32(
                false, afrag[q], false, b, (short)0, o, false, false);
        }
        #pragma unroll
        for (int v = 0; v < 8; ++v) {
            yrow0[(size_t)(v + roff) * D_OUT + n0 + m] = ALPHA * o[v];
        }
    }
}

extern "C" void kernel_launch(void* const* d_in, const int* in_sizes, int n_in,
                              void* d_out, int out_size, void* d_ws, size_t ws_size,
                              hipStream_t stream) {
    const float* x  = (const float*)d_in[0];
    const float* ht = (const float*)d_in[1];
    const float* Wa = (const float*)d_in[2];
    const float* Wb = (const float*)d_in[3];
    const float* Wc = (const float*)d_in[4];
    const float* Wd = (const float*)d_in[5];
    float* y = (float*)d_out;

    dim3 grid(NROWS / 16);   // 1024 blocks, one 16-row WMMA M-tile each
    dim3 block(256);         // 8 wave32
    ssm_lora_fused<<<grid, block, 0, stream>>>(x, ht, Wa, Wb, Wc, Wd, y);
}